// QParam_86131274154064
// MI455X (gfx1250) — compile-verified
//
#include <hip/hip_runtime.h>
#include <stdint.h>

#define QMAX_F 15.5f
#define NCODES 256

// ---------- CDNA5 async global->LDS feature detection (device side only) ----
#if defined(__gfx1250__) && __has_builtin(__builtin_amdgcn_global_load_async_to_lds_b32)
#define HAVE_ASYNC_LDS 1
#else
#define HAVE_ASYNC_LDS 0
#endif

#if HAVE_ASYNC_LDS
// Builtin signature (from hipcc diagnostic): (AS1 int*, AS3 int*, Ii, Ii)
typedef __attribute__((address_space(1))) int* as1_iptr;
typedef __attribute__((address_space(3))) int* as3_iptr;
#endif

__device__ __forceinline__ void wait_async_all() {
#if defined(__gfx1250__)
#if __has_builtin(__builtin_amdgcn_s_wait_asynccnt)
  __builtin_amdgcn_s_wait_asynccnt(0);
#else
  asm volatile("s_wait_asynccnt 0x0" ::: "memory");
#endif
#endif
}

// ---------------- Kernel 0: init workspace scalar (every call) --------------
__global__ void k_init(unsigned int* ws) {
  if (threadIdx.x == 0 && blockIdx.x == 0) ws[0] = 0u;  // bits of 0.0f
}

// ---------------- Kernel 1: abs-max reduction -------------------------------
// All |x| are >= 0, so the f32 bit pattern is order-preserving as u32 and a
// plain global_atomic_max_u32 implements the float max.
__global__ void k_absmax(const float* __restrict__ x, int n,
                         unsigned int* __restrict__ ws) {
  const int tid = blockIdx.x * blockDim.x + threadIdx.x;
  const int stride = gridDim.x * blockDim.x;
  const int n4 = n >> 2;
  const float4* __restrict__ x4 = (const float4*)x;

  float m = 0.0f;
  for (int i = tid; i < n4; i += stride) {
    if (i + stride < n4) __builtin_prefetch(&x4[i + stride], 0, 0);
    float4 v = x4[i];
    m = fmaxf(m, fmaxf(fmaxf(fabsf(v.x), fabsf(v.y)),
                       fmaxf(fabsf(v.z), fabsf(v.w))));
  }
  for (int i = (n4 << 2) + tid; i < n; i += stride) m = fmaxf(m, fabsf(x[i]));

  // wave32 butterfly reduction
#pragma unroll
  for (int off = 16; off > 0; off >>= 1)
    m = fmaxf(m, __shfl_xor(m, off, 32));

  if ((threadIdx.x & 31) == 0) atomicMax(ws, __float_as_uint(m));
}

// ---------------- nearest code via branchless lower_bound -------------------
// pos = min(#codes < q, 255) (in [0,255]); then compare the two bracketing codes.
// Tie -> lower index (matches jnp.argmin first-match semantics) via "<=".
__device__ __forceinline__ float nearest_code(const float* __restrict__ p,
                                              float q) {
  int pos = 0;
#pragma unroll
  for (int step = NCODES / 2; step >= 1; step >>= 1) {
    if (p[pos + step - 1] < q) pos += step;
  }
  float xh;
  if (pos == 0) {
    xh = p[0];
  } else if (pos == NCODES) {
    xh = p[NCODES - 1];
  } else {
    float lo = p[pos - 1], hi = p[pos];
    xh = ((q - lo) <= (hi - q)) ? lo : hi;
  }
  // Reproduce the reference's STE arithmetic exactly: (xhard - q) + q in f32.
  return (xh - q) + q;
}

// ---------------- Kernel 2: quantize + nearest-codeword map -----------------
// blockDim.x must be NCODES (256): each thread async-copies one code to LDS.
__global__ void k_quant(const float* __restrict__ x,
                        const float* __restrict__ plist,
                        const unsigned int* __restrict__ ws,
                        float* __restrict__ out, int n) {
  __shared__ float p[NCODES];
  const int t = threadIdx.x;

#if HAVE_ASYNC_LDS
  // gfx1250 async path: GLOBAL_LOAD_ASYNC_TO_LDS_B32 (tracked by ASYNCcnt)
  {
    float* gp = const_cast<float*>(plist + t);
    __builtin_amdgcn_global_load_async_to_lds_b32((as1_iptr)gp,
                                                  (as3_iptr)(p + t), 0, 0);
  }
  wait_async_all();
#else
  p[t] = plist[t];
#endif
  __syncthreads();

  const float scale = __uint_as_float(*ws) / QMAX_F;

  const int tid = blockIdx.x * blockDim.x + t;
  const int stride = gridDim.x * blockDim.x;
  const int n4 = n >> 2;
  const float4* __restrict__ x4 = (const float4*)x;
  float4* __restrict__ o4 = (float4*)out;

  for (int i = tid; i < n4; i += stride) {
    if (i + stride < n4) __builtin_prefetch(&x4[i + stride], 0, 0);
    float4 v = x4[i];
    float4 r;
    r.x = nearest_code(p, fminf(fmaxf(v.x / scale, -QMAX_F), QMAX_F));
    r.y = nearest_code(p, fminf(fmaxf(v.y / scale, -QMAX_F), QMAX_F));
    r.z = nearest_code(p, fminf(fmaxf(v.z / scale, -QMAX_F), QMAX_F));
    r.w = nearest_code(p, fminf(fmaxf(v.w / scale, -QMAX_F), QMAX_F));
    o4[i] = r;
  }
  for (int i = (n4 << 2) + tid; i < n; i += stride) {
    float q = fminf(fmaxf(x[i] / scale, -QMAX_F), QMAX_F);
    out[i] = nearest_code(p, q);
  }
}

// ---------------------------- launcher --------------------------------------
extern "C" void kernel_launch(void* const* d_in, const int* in_sizes, int n_in,
                              void* d_out, int out_size, void* d_ws,
                              size_t ws_size, hipStream_t stream) {
  const float* x = (const float*)d_in[0];
  const float* plist = (const float*)d_in[1];
  float* out = (float*)d_out;
  unsigned int* ws = (unsigned int*)d_ws;
  const int n = in_sizes[0];

  const int threads = 256;
  int work4 = n / 4;
  if (work4 < 1) work4 = 1;
  int blocks = (work4 + threads - 1) / threads;
  if (blocks < 1) blocks = 1;
  int blocks_r = blocks > 1024 ? 1024 : blocks;  // reduction: cap atomics
  int blocks_q = blocks > 4096 ? 4096 : blocks;  // map: ~1 float4/thread

  k_init<<<1, 32, 0, stream>>>(ws);
  k_absmax<<<blocks_r, threads, 0, stream>>>(x, n, ws);
  k_quant<<<blocks_q, threads, 0, stream>>>(x, plist, ws, out, n);
}